// Attention_Temp_1468878815458
// MI455X (gfx1250) — compile-verified
//
#include <hip/hip_runtime.h>

typedef float v2f __attribute__((ext_vector_type(2)));
typedef float v4f __attribute__((ext_vector_type(4)));
typedef float v8f __attribute__((ext_vector_type(8)));

// Problem sizes (B=8192, S=8, D=96)
constexpr int BB    = 8192;
constexpr int SS    = 8;
constexpr int DD    = 96;
constexpr int ROWS  = BB * SS;        // 65536 rows of 96 floats
constexpr int TILES = ROWS / 16;      // 4096 tiles of 16 rows
constexpr int NBLK  = 512;            // 512 blocks * 8 waves = 4096 waves = 1 tile/wave

static_assert(ROWS % 16 == 0, "tile cover");

// ---------------------------------------------------------------------------
// Precompute: cv[e]     = sum_n Wv[n,e]                      -> ws[0..95]
//             poscv[s]  = (sum_j j*Wp[s,j] + bp[s]) * sum_e cv[e] -> ws[96..103]
// Folding pos into a single post-WMMA scalar add removes the per-chunk
// load->add->wmma dependency in the streaming kernel.
// ---------------------------------------------------------------------------
__global__ void precompute_kernel(const float* __restrict__ Wp,
                                  const float* __restrict__ bp,
                                  const float* __restrict__ Wv,
                                  float* __restrict__ ws) {
  __shared__ float scv[DD];
  const int t = threadIdx.x;
  if (t < DD) {
    float s = 0.0f;
    for (int n = 0; n < DD; ++n) s += Wv[n * DD + t];
    scv[t] = s;
    ws[t]  = s;
  }
  __syncthreads();
  if (t < SS) {
    float cs = 0.0f;
    for (int e = 0; e < DD; ++e) cs += scv[e];
    float p = bp[t];
    for (int j = 0; j < SS; ++j) p += (float)j * Wp[t * SS + j];
    ws[DD + t] = p * cs;
  }
}

// ---------------------------------------------------------------------------
// Streaming kernel: out[row, :] = dot(x[row,:], cv) + poscv[row%8]
// One 16-row tile per wave; dot via 24x V_WMMA_F32_16X16X4_F32:
//   A[m,k] = cv[kb+k]          (broadcast over M, hoisted from LDS)
//   B[k,n] = x[row_n, kb+k]    (lane L: global_load_b64 at row*96 + kb + 2h)
//   C[m,n] = dot[row_n]        -> every lane holds dot[row_{L%16}] in c[0]
// WMMAs consume the loads directly (no intervening VALU), so all 24 b64
// loads can be in flight before the first matrix op.
// ---------------------------------------------------------------------------
__global__ void __launch_bounds__(256)
attn_stream_kernel(const float* __restrict__ x,
                   const float* __restrict__ ws,
                   float* __restrict__ out) {
  __shared__ float lds[104];
  const int tid = threadIdx.x;
  if (tid < 104) lds[tid] = ws[tid];
  __syncthreads();

  const int lane = tid & 31;
  const int h    = lane >> 4;   // half: 0 -> K{0,1}/cols 0..47, 1 -> K{2,3}/cols 48..95
  const int n    = lane & 15;   // row within 16-row tile
  const float poscv = lds[DD + (n & 7)];  // s = row % 8 == n % 8 (tiles 8-aligned)

  // Hoist all 24 A-operand chunks (cv) out of the tile loop.
  v2f a[24];
#pragma unroll
  for (int c = 0; c < 24; ++c) {
    a[c].x = lds[4 * c + 2 * h];
    a[c].y = lds[4 * c + 2 * h + 1];
  }

  const int wave   = blockIdx.x * (blockDim.x >> 5) + (tid >> 5);
  const int nwaves = gridDim.x * (blockDim.x >> 5);

  for (int tile = wave; tile < TILES; tile += nwaves) {  // exactly one pass
    const int row = tile * 16 + n;
    const float* gp = x + (size_t)row * DD + 2 * h;

    // 24 x global_load_b64 with immediate offsets — all independent.
    v2f bmat[24];
#pragma unroll
    for (int c = 0; c < 24; ++c) {
      bmat[c] = *(const v2f*)(gp + 4 * c);
    }

    // 96-wide dot via 24 fp32 WMMAs, loads feed WMMA directly.
    v8f acc = {};
#pragma unroll
    for (int c = 0; c < 24; ++c) {
      acc = __builtin_amdgcn_wmma_f32_16x16x4_f32(
          /*neg_a=*/false, a[c], /*neg_b=*/false, bmat[c],
          /*c_mod=*/(short)0, acc, /*reuse_a=*/false, /*reuse_b=*/false);
    }

    // Every lane holds dot[row_{L%16}] in acc[0]; add the folded pos term
    // and splat-store its 48-column half of the row: 12 x global_store_b128.
    const float v = acc[0] + poscv;
    const v4f sp = {v, v, v, v};
    float* op = out + (size_t)row * DD + 48 * h;
#pragma unroll
    for (int i = 0; i < 12; ++i) {
      *(v4f*)(op + 4 * i) = sp;
    }
  }
}

// ---------------------------------------------------------------------------
// Host launcher
// ---------------------------------------------------------------------------
extern "C" void kernel_launch(void* const* d_in, const int* in_sizes, int n_in,
                              void* d_out, int out_size, void* d_ws, size_t ws_size,
                              hipStream_t stream) {
  const float* x  = (const float*)d_in[0];   // (B,1,S,D) fp32
  const float* Wp = (const float*)d_in[1];   // (8,8)
  const float* bp = (const float*)d_in[2];   // (8,)
  const float* Wv = (const float*)d_in[3];   // (96,96)
  // d_in[4] (Wk), d_in[5] (Wq): the softmax column-sums to 1 exactly, so the
  // Q/K path cancels out of the reference math entirely.

  float* ws  = (float*)d_ws;                 // needs 104 floats
  float* out = (float*)d_out;                // (B,S,D) fp32

  precompute_kernel<<<1, 128, 0, stream>>>(Wp, bp, Wv, ws);
  attn_stream_kernel<<<NBLK, 256, 0, stream>>>(x, ws, out);
}